// GINLayer_39273180954647
// MI455X (gfx1250) — compile-verified
//
#include <hip/hip_runtime.h>
#include <hip/hip_bf16.h>

typedef float v2f __attribute__((ext_vector_type(2)));
typedef float v8f __attribute__((ext_vector_type(8)));

#define DCOL 256
#define SCAN_THREADS 1024

// ---------------------------------------------------------------------------
// CSR build step 1: zero the degree counters
// ---------------------------------------------------------------------------
__global__ void __launch_bounds__(256) gin_zero(int* __restrict__ deg, int N) {
    int i = blockIdx.x * blockDim.x + threadIdx.x;
    if (i < N) deg[i] = 0;
}

// ---------------------------------------------------------------------------
// CSR build step 2: histogram of destination nodes (integer atomics only)
// ---------------------------------------------------------------------------
__global__ void __launch_bounds__(256) gin_hist(const long long* __restrict__ ei,
                                                int* __restrict__ deg, int E) {
    int e = blockIdx.x * blockDim.x + threadIdx.x;
    if (e < E) atomicAdd(&deg[(int)ei[E + e]], 1);
}

// ---------------------------------------------------------------------------
// CSR build step 3: single-block exclusive scan (N <= SCAN_THREADS * chunk).
// Hillis-Steele over per-thread chunk sums in LDS, then write offsets+cursors.
// ---------------------------------------------------------------------------
__global__ void __launch_bounds__(SCAN_THREADS) gin_scan(const int* __restrict__ deg,
                                                         int* __restrict__ off,
                                                         int* __restrict__ cursor, int N) {
    __shared__ int partial[SCAN_THREADS];
    const int t = threadIdx.x;
    const int chunk = (N + SCAN_THREADS - 1) / SCAN_THREADS;
    const int beg = t * chunk;
    const int end = min(beg + chunk, N);
    int s = 0;
    for (int i = beg; i < end; ++i) s += deg[i];
    partial[t] = s;
    __syncthreads();
    for (int d = 1; d < SCAN_THREADS; d <<= 1) {
        int v = partial[t];
        int add = (t >= d) ? partial[t - d] : 0;
        __syncthreads();
        partial[t] = v + add;
        __syncthreads();
    }
    int run = (t == 0) ? 0 : partial[t - 1];   // exclusive prefix of this chunk
    for (int i = beg; i < end; ++i) {
        off[i] = run;
        cursor[i] = run;
        run += deg[i];
    }
    if (t == SCAN_THREADS - 1) off[N] = run;   // == E
}

// ---------------------------------------------------------------------------
// CSR build step 4: bucket-fill sources by destination
// ---------------------------------------------------------------------------
__global__ void __launch_bounds__(256) gin_fill(const long long* __restrict__ ei,
                                                int* __restrict__ cursor,
                                                int* __restrict__ csr_src, int E) {
    int e = blockIdx.x * blockDim.x + threadIdx.x;
    if (e < E) {
        int d = (int)ei[E + e];
        int pos = atomicAdd(&cursor[d], 1);
        csr_src[pos] = (int)ei[e];
    }
}

// ---------------------------------------------------------------------------
// 256x256 transpose (LDS tiled): Wt[n][k] = W[k][n].  Lets the GEMM load B
// fragments as contiguous b64 pairs instead of two 1KB-strided b32 loads.
// ---------------------------------------------------------------------------
__global__ void __launch_bounds__(256) gin_transpose(const float* __restrict__ W,
                                                     float* __restrict__ Wt) {
    __shared__ float tile[16][17];
    const int bx = blockIdx.x & 15;
    const int by = blockIdx.x >> 4;
    const int tx = threadIdx.x & 15;
    const int ty = threadIdx.x >> 4;
    tile[ty][tx] = W[(by * 16 + ty) * DCOL + bx * 16 + tx];
    __syncthreads();
    // Wt[bx*16+ty][by*16+tx] = W[by*16+tx][bx*16+ty]
    Wt[(bx * 16 + ty) * DCOL + by * 16 + tx] = tile[tx][ty];
}

// ---------------------------------------------------------------------------
// Aggregation as a GATHER: h[i] = x[i] + sum_{e: dst==i} x[src_e].
// 64 lanes x float4 per node, 4 nodes per 256-thread block.  All reads are
// coalesced b128 loads from the L2-resident x; zero floating-point atomics.
// ---------------------------------------------------------------------------
__global__ void __launch_bounds__(256) gin_aggregate(const float* __restrict__ x,
                                                     const int* __restrict__ off,
                                                     const int* __restrict__ csr_src,
                                                     float* __restrict__ h, int N) {
    int node = blockIdx.x * 4 + (threadIdx.x >> 6);
    if (node >= N) return;
    int c = (threadIdx.x & 63) << 2;

    const int beg = off[node];
    const int end = off[node + 1];

    float4 s0 = *(const float4*)(x + (size_t)node * DCOL + c);   // self term (eps=0)
    float4 s1 = make_float4(0.f, 0.f, 0.f, 0.f);

    int e = beg;
    for (; e + 1 < end; e += 2) {
        int a = csr_src[e];
        int b = csr_src[e + 1];
        float4 va = *(const float4*)(x + (size_t)a * DCOL + c);
        float4 vb = *(const float4*)(x + (size_t)b * DCOL + c);
        s0.x += va.x; s0.y += va.y; s0.z += va.z; s0.w += va.w;
        s1.x += vb.x; s1.y += vb.y; s1.z += vb.z; s1.w += vb.w;
    }
    if (e < end) {
        int a = csr_src[e];
        float4 va = *(const float4*)(x + (size_t)a * DCOL + c);
        s0.x += va.x; s0.y += va.y; s0.z += va.z; s0.w += va.w;
    }
    s0.x += s1.x; s0.y += s1.y; s0.z += s1.z; s0.w += s1.w;
    *(float4*)(h + (size_t)node * DCOL + c) = s0;
}

// ---------------------------------------------------------------------------
// Out[M,256] = act(H[M,256] @ W[256,256] + bias), W given TRANSPOSED (Wt).
// One wave32 computes a 16(M) x 64(N) strip with V_WMMA_F32_16X16X4_F32.
//   A frag (16x4 f32, v2f): lane<16 -> row M=lane, K={0,1}; lane>=16 -> K={2,3}
//   B frag (4x16  f32, v2f): lanes 0-15 col N=lane K={0,1}; lanes 16-31 K={2,3}
//     -> from Wt: contiguous pair Wt[n][k0+2hi], Wt[n][k0+2hi+1]  (b64 load)
//   C/D (16x16 f32, v8f): element (vgpr r, lane L) = row r + 8*(L>>4), col L&15
// Per k-step: 1 b64 A load + 4 b64 B loads for 4 WMMAs.
// ---------------------------------------------------------------------------
__global__ void __launch_bounds__(256) gin_gemm(const float* __restrict__ H,
                                                const float* __restrict__ Wt,
                                                const float* __restrict__ bias,
                                                float* __restrict__ Out,
                                                int M, int doRelu) {
    const int lane  = threadIdx.x & 31;
    const int wid   = blockIdx.x * (blockDim.x >> 5) + (threadIdx.x >> 5);
    const int mTile = wid >> 2;           // 16 rows per tile
    const int nBase = (wid & 3) << 6;     // 64 cols per wave
    if (mTile * 16 >= M) return;          // wave-uniform: EXEC stays all-ones

    const int lo = lane & 15;
    const int hi = lane >> 4;

    const float* hrow = H + (size_t)(mTile * 16 + lo) * DCOL + hi * 2;
    const float* wrow = Wt + (size_t)(nBase + lo) * DCOL + hi * 2;  // +j*16*DCOL per frag

    v8f acc[4];
#pragma unroll
    for (int j = 0; j < 4; ++j) {
        float bv = bias[nBase + j * 16 + lo];   // whole lane shares one column
        acc[j] = (v8f){bv, bv, bv, bv, bv, bv, bv, bv};
    }

#pragma unroll 4
    for (int k0 = 0; k0 < DCOL; k0 += 4) {
        v2f a = *(const v2f*)(hrow + k0);                 // A[M=lo][k0+2hi..+1]
#pragma unroll
        for (int j = 0; j < 4; ++j) {
            v2f b = *(const v2f*)(wrow + (size_t)j * 16 * DCOL + k0);  // Wt[n][k0+2hi..+1]
            acc[j] = __builtin_amdgcn_wmma_f32_16x16x4_f32(
                false, a, false, b, (short)0, acc[j], false, false);
        }
    }

#pragma unroll
    for (int j = 0; j < 4; ++j) {
        int n = nBase + j * 16 + lo;
#pragma unroll
        for (int r = 0; r < 8; ++r) {
            float v = acc[j][r];
            if (doRelu) v = v > 0.0f ? v : 0.0f;
            Out[(size_t)(mTile * 16 + hi * 8 + r) * DCOL + n] = v;
        }
    }
}

// ---------------------------------------------------------------------------
extern "C" void kernel_launch(void* const* d_in, const int* in_sizes, int n_in,
                              void* d_out, int out_size, void* d_ws, size_t ws_size,
                              hipStream_t stream) {
    const float*     x  = (const float*)d_in[0];
    const long long* ei = (const long long*)d_in[1];   // [2, E] int64
    const float*     W1 = (const float*)d_in[2];
    const float*     b1 = (const float*)d_in[3];
    const float*     W2 = (const float*)d_in[4];
    const float*     b2 = (const float*)d_in[5];

    const int N = in_sizes[0] / DCOL;   // 10000
    const int E = in_sizes[1] / 2;      // 320000

    // workspace layout
    float* h   = (float*)d_ws;                         // [N,256]  x + agg
    float* h1  = h + (size_t)N * DCOL;                 // [N,256]  relu(h W1 + b1)
    float* Wt1 = h1 + (size_t)N * DCOL;                // [256,256] W1^T
    float* Wt2 = Wt1 + DCOL * DCOL;                    // [256,256] W2^T
    int*   deg = (int*)(Wt2 + DCOL * DCOL);            // [N]
    int*   off = deg + N;                              // [N+1]
    int*   cur = off + N + 1;                          // [N]
    int*   csr = cur + N;                              // [E]
    float* out = (float*)d_out;

    // 0) transpose weights (256KB each; negligible, enables b64 B-frag loads)
    gin_transpose<<<256, 256, 0, stream>>>(W1, Wt1);
    gin_transpose<<<256, 256, 0, stream>>>(W2, Wt2);

    // 1-4) build CSR by destination (integer atomics only)
    gin_zero<<<(N + 255) / 256, 256, 0, stream>>>(deg, N);
    gin_hist<<<(E + 255) / 256, 256, 0, stream>>>(ei, deg, E);
    gin_scan<<<1, SCAN_THREADS, 0, stream>>>(deg, off, cur, N);
    gin_fill<<<(E + 255) / 256, 256, 0, stream>>>(ei, cur, csr, E);

    // 5) gather-aggregate: h = x + sum of neighbor rows (no FP atomics)
    gin_aggregate<<<(N + 3) / 4, 256, 0, stream>>>(x, off, csr, h, N);

    // 6) h1 = relu(h @ W1 + b1)
    int mTiles = (N + 15) / 16;               // 625
    int waves  = mTiles * 4;                  // one wave per 16x64 strip
    int blocks = (waves + 7) / 8;             // 8 waves (256 thr) per block
    gin_gemm<<<blocks, 256, 0, stream>>>(h, Wt1, b1, h1, N, 1);

    // 7) out = h1 @ W2 + b2
    gin_gemm<<<blocks, 256, 0, stream>>>(h1, Wt2, b2, out, N, 0);
}